// BandwidthMask_13013750906886
// MI455X (gfx1250) — compile-verified
//
#include <hip/hip_runtime.h>

// ---------------------------------------------------------------------------
// Segment (per-source-node) softmax with temperature over edge weights.
//   out_e = exp((x_e - max_{row(e)}) / t) / (sum_{row(e)} exp(...) + 1e-16)
//
// E = 16M edges (row sorted ascending), N = 500K nodes. HBM-bound
// (~640MB traffic vs ~60 MFLOP). Wave32 segmented scans (ballot/head-flag
// form) cut global atomics ~16-32x; pass 1 compacts int64 rows to int32 in
// scratch so later passes read half the index bytes; pass 3 is float4/int4
// vectorized.
// ---------------------------------------------------------------------------

#define WAVE 32

// Order-preserving float <-> uint key: enables native GLOBAL_ATOMIC_MAX_U32.
__device__ __forceinline__ unsigned fkey(float f) {
    unsigned u = __float_as_uint(f);
    return (u & 0x80000000u) ? ~u : (u | 0x80000000u);
}
__device__ __forceinline__ float funkey(unsigned k) {
    unsigned u = (k & 0x80000000u) ? (k & 0x7fffffffu) : ~k;
    return __uint_as_float(u);
}

// Run-head detection + segmented-scan helpers (wave32).
// Returns the lane index where this lane's run starts, and the ballot mask.
__device__ __forceinline__ int run_start(int r, int lane, unsigned& hb32) {
    const int rprev = __shfl_up(r, 1, WAVE);
    const bool head = (lane == 0) || (rprev != r);
    hb32 = (unsigned)__ballot(head);                    // bit0 always set
    const unsigned le_mask = (lane == WAVE - 1) ? 0xffffffffu
                                                : ((1u << (lane + 1)) - 1u);
    return 31 - __clz(hb32 & le_mask);                  // highest head <= lane
}
__device__ __forceinline__ bool run_last(int lane, unsigned hb32) {
    return (lane == WAVE - 1) ? true : (((hb32 >> (lane + 1)) & 1u) != 0u);
}

// Pass 0: zero per-node state. N only known on-device.
__global__ __launch_bounds__(256)
void seg_init_kernel(unsigned* __restrict__ nodebase, const int* __restrict__ nptr) {
    const int n = *nptr;                        // low word of int32/int64 scalar
    unsigned* kmax = nodebase;                  // [0, n)
    float*    ssum = (float*)(nodebase + n);    // [n, 2n)
    for (int i = blockIdx.x * blockDim.x + threadIdx.x; i < n;
         i += gridDim.x * blockDim.x) {
        kmax[i] = 0u;                           // below any real key
        ssum[i] = 0.0f;
    }
}

// Pass 1: per-segment max via wave32 segmented max-scan + boundary atomics.
// Also compacts int64 row ids to int32 scratch (if provided).
__global__ __launch_bounds__(256)
void seg_max_kernel(const long long* __restrict__ row64,
                    const float* __restrict__ x,
                    unsigned* __restrict__ nodebase,
                    int* __restrict__ row32,          // nullable
                    int E) {
    unsigned* kmax = nodebase;
    const int lane = threadIdx.x & (WAVE - 1);
    const int wid  = (blockIdx.x * blockDim.x + threadIdx.x) >> 5;
    const int nwv  = (gridDim.x * blockDim.x) >> 5;

    for (long long base = (long long)wid * WAVE; base < E;
         base += (long long)nwv * WAVE) {
        const long long i = base + lane;
        const bool valid = (i < E);
        int   r = valid ? (int)__builtin_nontemporal_load(&row64[i]) : -1;
        float v = valid ? __builtin_nontemporal_load(&x[i]) : -3.402823466e+38f;
        if (row32 && valid) row32[i] = r;       // keep resident for passes 2-3

        unsigned hb;
        const int start = run_start(r, lane, hb);
        #pragma unroll
        for (int off = 1; off < WAVE; off <<= 1) {
            const float vv = __shfl_up(v, off, WAVE);
            if (lane >= start + off) v = fmaxf(v, vv);
        }
        if (valid && run_last(lane, hb)) atomicMax(&kmax[r], fkey(v));
    }
}

// Pass 2: z = exp((x - seg_max[row]) / t) -> out; segmented sum-scan,
// boundary lanes atomicAdd (native GLOBAL_ATOMIC_ADD_F32).
__global__ __launch_bounds__(256)
void seg_expsum_kernel(const long long* __restrict__ row64,
                       const int* __restrict__ row32,   // nullable
                       const float* __restrict__ x,
                       unsigned* __restrict__ nodebase,
                       const int* __restrict__ nptr,
                       const int* __restrict__ tptr,
                       float* __restrict__ z,
                       int E) {
    const int n = *nptr;
    unsigned* kmax = nodebase;
    float*    ssum = (float*)(nodebase + n);
    const float inv_t = 1.0f / (float)(*tptr);

    const int lane = threadIdx.x & (WAVE - 1);
    const int wid  = (blockIdx.x * blockDim.x + threadIdx.x) >> 5;
    const int nwv  = (gridDim.x * blockDim.x) >> 5;

    for (long long base = (long long)wid * WAVE; base < E;
         base += (long long)nwv * WAVE) {
        const long long i = base + lane;
        const bool valid = (i < E);
        int r = -1;
        if (valid) r = row32 ? row32[i] : (int)__builtin_nontemporal_load(&row64[i]);
        const float v = valid ? __builtin_nontemporal_load(&x[i]) : 0.0f;

        float zz = 0.0f;
        if (valid) {
            const float m = funkey(kmax[r]);    // L2-hot, wave-coherent gather
            zz = __expf((v - m) * inv_t);       // v_exp_f32
            z[i] = zz;                          // RT store: reused in pass 3
        }

        unsigned hb;
        const int start = run_start(r, lane, hb);
        #pragma unroll
        for (int off = 1; off < WAVE; off <<= 1) {
            const float s = __shfl_up(zz, off, WAVE);
            if (lane >= start + off) zz += s;
        }
        if (valid && run_last(lane, hb)) atomicAdd(&ssum[r], zz);
    }
}

// Pass 3 (vectorized): out = z / (seg_sum[row] + 1e-16), 4 edges/lane with
// b128 loads/stores. In place over d_out.
__global__ __launch_bounds__(256)
void seg_norm4_kernel(const int* __restrict__ row32,
                      const unsigned* __restrict__ nodebase,
                      const int* __restrict__ nptr,
                      float* z_out,
                      int E) {
    const int n = *nptr;
    const float* ssum = (const float*)(nodebase + n);
    const int nq = E >> 2;
    const long long tid    = blockIdx.x * (long long)blockDim.x + threadIdx.x;
    const long long stride = (long long)gridDim.x * blockDim.x;

    const int4*  r4p = (const int4*)row32;
    float4*      z4p = (float4*)z_out;
    for (long long q = tid; q < nq; q += stride) {
        const int4   r4 = r4p[q];
        const float4 z4 = z4p[q];
        float4 o;
        o.x = __fdividef(z4.x, ssum[r4.x] + 1e-16f);
        o.y = __fdividef(z4.y, ssum[r4.y] + 1e-16f);
        o.z = __fdividef(z4.z, ssum[r4.z] + 1e-16f);
        o.w = __fdividef(z4.w, ssum[r4.w] + 1e-16f);
        z4p[q] = o;
    }
    for (long long i = (long long)nq * 4 + tid; i < E; i += stride) {  // tail
        z_out[i] = __fdividef(z_out[i], ssum[row32[i]] + 1e-16f);
    }
}

// Pass 3 (fallback, no int32 scratch): scalar with int64 rows.
__global__ __launch_bounds__(256)
void seg_norm_kernel(const long long* __restrict__ row64,
                     const unsigned* __restrict__ nodebase,
                     const int* __restrict__ nptr,
                     float* z_out,
                     int E) {
    const int n = *nptr;
    const float* ssum = (const float*)(nodebase + n);
    for (long long i = blockIdx.x * (long long)blockDim.x + threadIdx.x; i < E;
         i += (long long)gridDim.x * blockDim.x) {
        const int r = (int)__builtin_nontemporal_load(&row64[i]);
        z_out[i] = __fdividef(z_out[i], ssum[r] + 1e-16f);
    }
}

extern "C" void kernel_launch(void* const* d_in, const int* in_sizes, int n_in,
                              void* d_out, int out_size, void* d_ws, size_t ws_size,
                              hipStream_t stream) {
    // setup_inputs order: edge_index [2,E] int64, bandwidth [E] f32,
    //                     num_nodes scalar, t scalar
    const long long* row64 = (const long long*)d_in[0];   // edge_index[0] = first E
    const float*     bw    = (const float*)d_in[1];
    const int*       nptr  = (const int*)d_in[2];
    const int*       tptr  = (const int*)d_in[3];

    const int E   = in_sizes[1];
    float*    out = (float*)d_out;

    // Scratch layout: [optional row32: E*4B][kmax: n*4B][ssum: n*4B].
    // Node arrays need 8*N bytes; reserve 16MB (N up to 2M) when deciding
    // whether the int32 row compaction fits. Deterministic host-side branch.
    const size_t row32_bytes  = (size_t)E * sizeof(int);
    const size_t node_reserve = (size_t)16 << 20;
    const bool   use_row32    = ws_size >= row32_bytes + node_reserve;

    int*      row32    = use_row32 ? (int*)d_ws : nullptr;
    unsigned* nodebase = use_row32 ? (unsigned*)((char*)d_ws + row32_bytes)
                                   : (unsigned*)d_ws;

    const dim3 blk(256);
    int blocks = (E + 255) / 256;
    if (blocks > 32768) blocks = 32768;                   // grid-stride covers rest

    seg_init_kernel  <<<4096,   blk, 0, stream>>>(nodebase, nptr);
    seg_max_kernel   <<<blocks, blk, 0, stream>>>(row64, bw, nodebase, row32, E);
    seg_expsum_kernel<<<blocks, blk, 0, stream>>>(row64, row32, bw, nodebase,
                                                  nptr, tptr, out, E);
    if (use_row32) {
        seg_norm4_kernel<<<blocks, blk, 0, stream>>>(row32, nodebase, nptr, out, E);
    } else {
        seg_norm_kernel <<<blocks, blk, 0, stream>>>(row64, nodebase, nptr, out, E);
    }
}